// Net_86474871538316
// MI455X (gfx1250) — compile-verified
//
#include <hip/hip_runtime.h>
#include <cstdint>
#include <cstddef>

#define NN 50000
#define EE 800000
#define ETOT (EE + NN)
#define KF 128          // inner dim for every GEMM in this net
#define HH 4
#define NEG_SLOPE 0.2f
#define RT 4            // 16-row tiles per block -> 64 rows/block
#define LDS_STRIDE 132  // floats; 528B rows: 16B-aligned chunks, bank-conflict-free

typedef __attribute__((ext_vector_type(2))) float v2f;
typedef __attribute__((ext_vector_type(8))) float v8f;

// ---------------------------------------------------------------------------
// CDNA5 async copy: 16 bytes global -> LDS, tracked with ASYNCcnt.
// ---------------------------------------------------------------------------
__device__ __forceinline__ void async_copy16(unsigned lds_off, const float* gptr) {
  asm volatile("global_load_async_to_lds_b128 %0, %1, off"
               :: "v"(lds_off), "v"((unsigned long long)(uintptr_t)gptr)
               : "memory");
}
__device__ __forceinline__ void async_wait0() {
  asm volatile("s_wait_asynccnt 0x0" ::: "memory");
}

// ---------------------------------------------------------------------------
// GEMM: out[N,M] = A[N,128] @ W[128,M] (+ bias), fp32 WMMA 16x16x4.
// Block = 8 waves, 64 rows. B tile register-cached per wave (32 x v2f);
// A tile staged in LDS via async-to-LDS; inner loop = ds_load_b64 + v_wmma.
// ---------------------------------------------------------------------------
__global__ __launch_bounds__(256) void gemm_wmma_k(const float* __restrict__ A,
                                                   const float* __restrict__ W,
                                                   const float* __restrict__ bias,
                                                   float* __restrict__ out, int M) {
  __shared__ float lds[16 * RT * LDS_STRIDE];
  const int tid = threadIdx.x;
  const int n0  = blockIdx.x * (16 * RT);

  // ---- async stage of the 64x128 fp32 A tile (16B per issue) ----
  {
    const unsigned base = (unsigned)(uintptr_t)(void*)lds;  // LDS byte offset (addr[31:0])
    for (int i = tid; i < 16 * RT * 32; i += 256) {
      const int row  = i >> 5, c4 = i & 31;
      int grow = n0 + row;
      if (grow >= NN) grow = NN - 1;                        // clamp; stores are masked
      const unsigned loff = base + (unsigned)((row * LDS_STRIDE + c4 * 4) * 4);
      async_copy16(loff, A + (size_t)grow * KF + c4 * 4);
    }
    async_wait0();
  }
  __syncthreads();

  const int lane = tid & 31;
  const int wave = tid >> 5;
  const int half = lane >> 4;   // K-pair selector
  const int idx  = lane & 15;   // row (for A) / col (for B, C/D)
  const int Mt   = (M + 15) >> 4;

  if (wave >= Mt) return;       // after barrier; waves idle for M<128
  const int  m0   = wave << 4;
  const int  col  = m0 + idx;
  const bool ok   = col < M;
  const int  colc = ok ? col : (M - 1);

  // ---- register-cache B tile: B[kk, col] for all 32 K-steps ----
  v2f breg[32];
#pragma unroll
  for (int k0 = 0; k0 < 32; ++k0) {
    const int kk = 4 * k0 + 2 * half;
    const float w0 = W[(size_t)kk * M + colc];
    const float w1 = W[(size_t)(kk + 1) * M + colc];
    breg[k0].x = ok ? w0 : 0.f;
    breg[k0].y = ok ? w1 : 0.f;
  }
  const float bv = (ok && bias) ? bias[col] : 0.f;

  // ---- 4 row tiles x 32 chained fp32 WMMAs ----
  for (int r = 0; r < RT; ++r) {
    const float* ap = &lds[(r * 16 + idx) * LDS_STRIDE + 2 * half];
    v8f acc = {};
#pragma unroll
    for (int k0 = 0; k0 < 32; ++k0) {
      const v2f a = *reinterpret_cast<const v2f*>(ap + 4 * k0);  // A[row, kk..kk+1]
      acc = __builtin_amdgcn_wmma_f32_16x16x4_f32(false, a, false, breg[k0],
                                                  (short)0, acc, false, false);
    }
    if (ok) {
#pragma unroll
      for (int rr = 0; rr < 8; ++rr) {
        const int row = n0 + r * 16 + rr + 8 * half;  // C/D: VGPR rr -> M = rr + 8*half
        if (row < NN) out[(size_t)row * M + col] = acc[rr] + bv;
      }
    }
  }
}

// ---------------------------------------------------------------------------
// Per-(node,head) attention coefficients: alpha = <h[n,head,:], a[head,:]>
// ---------------------------------------------------------------------------
__global__ void alpha_k(const float* __restrict__ h, const float* __restrict__ as,
                        const float* __restrict__ ad, float* __restrict__ asrc,
                        float* __restrict__ adst, int Cout, int Ctot) {
  const int t = blockIdx.x * blockDim.x + threadIdx.x;
  if (t >= NN * HH) return;
  const int n = t >> 2, hd = t & 3;
  const float* hp  = h  + (size_t)n * Ctot + hd * Cout;
  const float* asp = as + hd * Cout;
  const float* adp = ad + hd * Cout;
  float s = 0.f, d = 0.f;
  for (int c = 0; c < Cout; ++c) {
    const float v = hp[c];
    s += v * asp[c];
    d += v * adp[c];
  }
  asrc[t] = s;
  adst[t] = d;
}

__global__ void fill_u32_k(unsigned* __restrict__ p, unsigned v, int n) {
  const int t = blockIdx.x * blockDim.x + threadIdx.x;
  if (t < n) p[t] = v;
}

// order-preserving float<->uint transform for atomicMax on floats
__device__ __forceinline__ unsigned fenc(float f) {
  const unsigned u = __float_as_uint(f);
  return (u & 0x80000000u) ? ~u : (u | 0x80000000u);
}
__device__ __forceinline__ float fdec(unsigned u) {
  return (u & 0x80000000u) ? __uint_as_float(u & 0x7FFFFFFFu) : __uint_as_float(~u);
}

__device__ __forceinline__ void edge_sd(const int* __restrict__ ei, int e, int& s, int& d) {
  if (e < EE) { s = ei[e]; d = ei[EE + e]; }
  else        { s = d = e - EE; }  // self-loops appended
}

// pass 1: e = leakyrelu(asrc[src]+adst[dst]); segment max over dst (atomic)
__global__ void edge_max_k(const int* __restrict__ ei, const float* __restrict__ asrc,
                           const float* __restrict__ adst, unsigned* __restrict__ menc,
                           float* __restrict__ pbuf) {
  const int e = blockIdx.x * blockDim.x + threadIdx.x;
  if (e >= ETOT) return;
  int s, d; edge_sd(ei, e, s, d);
#pragma unroll
  for (int h = 0; h < HH; ++h) {
    float ev = asrc[s * HH + h] + adst[d * HH + h];
    ev = ev > 0.f ? ev : NEG_SLOPE * ev;
    pbuf[(size_t)e * HH + h] = ev;
    atomicMax(&menc[d * HH + h], fenc(ev));
  }
}

// pass 2: p = exp(e - m[dst]); segment sum over dst
__global__ void edge_expsum_k(const int* __restrict__ ei, const unsigned* __restrict__ menc,
                              float* __restrict__ denom, float* __restrict__ pbuf) {
  const int e = blockIdx.x * blockDim.x + threadIdx.x;
  if (e >= ETOT) return;
  int s, d; edge_sd(ei, e, s, d);
#pragma unroll
  for (int h = 0; h < HH; ++h) {
    const float pe = expf(pbuf[(size_t)e * HH + h] - fdec(menc[d * HH + h]));
    pbuf[(size_t)e * HH + h] = pe;
    atomicAdd(&denom[d * HH + h], pe);
  }
}

// pass 3: alpha = p / (denom[dst] + 1e-16)
__global__ void edge_norm_k(const int* __restrict__ ei, const float* __restrict__ denom,
                            float* __restrict__ pbuf) {
  const int e = blockIdx.x * blockDim.x + threadIdx.x;
  if (e >= ETOT) return;
  int s, d; edge_sd(ei, e, s, d);
#pragma unroll
  for (int h = 0; h < HH; ++h)
    pbuf[(size_t)e * HH + h] /= (denom[d * HH + h] + 1e-16f);
}

// pass 4: gat[dst, c] += alpha[e, head(c)] * h[src, c]   (block = one edge)
__global__ void edge_agg_k(const int* __restrict__ ei, const float* __restrict__ pbuf,
                           const float* __restrict__ hbuf, float* __restrict__ gat,
                           int Ctot, int Cout) {
  const int e = blockIdx.x;
  const int c = threadIdx.x;
  if (c >= Ctot) return;
  int s, d; edge_sd(ei, e, s, d);
  const int hd = c / Cout;
  const float w = pbuf[(size_t)e * HH + hd];
  atomicAdd(&gat[(size_t)d * Ctot + c], w * hbuf[(size_t)s * Ctot + c]);
}

// x' = elu(gat + b + lin)   (lin already carries its own bias from the GEMM)
__global__ void finalize12_k(const float* __restrict__ gat, const float* __restrict__ b,
                             const float* __restrict__ lin, float* __restrict__ xout) {
  const int t = blockIdx.x * blockDim.x + threadIdx.x;
  if (t >= NN * KF) return;
  const int f = t & 127;
  const float v = gat[t] + b[f] + lin[t];
  xout[t] = v > 0.f ? v : expm1f(v);
}

// out = mean_heads(gat3) + b3 + lin3
__global__ void finalize3_k(const float* __restrict__ gat, const float* __restrict__ b3,
                            const float* __restrict__ lin3, float* __restrict__ out) {
  const int t = blockIdx.x * blockDim.x + threadIdx.x;
  if (t >= NN * 10) return;
  const int n = t / 10, cls = t % 10;
  const float g = 0.25f * (gat[(size_t)n * 40 + cls]      + gat[(size_t)n * 40 + 10 + cls] +
                           gat[(size_t)n * 40 + 20 + cls] + gat[(size_t)n * 40 + 30 + cls]);
  out[t] = g + b3[cls] + lin3[(size_t)n * 10 + cls];
}

// ---------------------------------------------------------------------------
extern "C" void kernel_launch(void* const* d_in, const int* in_sizes, int n_in,
                              void* d_out, int out_size, void* d_ws, size_t ws_size,
                              hipStream_t stream) {
  (void)in_sizes; (void)n_in; (void)out_size; (void)ws_size;
  const float* x   = (const float*)d_in[0];
  const int*   ei  = (const int*)d_in[1];
  const float* W1  = (const float*)d_in[2];
  const float* a1s = (const float*)d_in[3];
  const float* a1d = (const float*)d_in[4];
  const float* b1  = (const float*)d_in[5];
  const float* l1W = (const float*)d_in[6];
  const float* l1b = (const float*)d_in[7];
  const float* W2  = (const float*)d_in[8];
  const float* a2s = (const float*)d_in[9];
  const float* a2d = (const float*)d_in[10];
  const float* b2  = (const float*)d_in[11];
  const float* l2W = (const float*)d_in[12];
  const float* l2b = (const float*)d_in[13];
  const float* W3  = (const float*)d_in[14];
  const float* a3s = (const float*)d_in[15];
  const float* a3d = (const float*)d_in[16];
  const float* b3  = (const float*)d_in[17];
  const float* l3W = (const float*)d_in[18];
  const float* l3b = (const float*)d_in[19];
  float* out = (float*)d_out;

  // workspace carve-up (floats)
  float*    A     = (float*)d_ws;                 // h buffer      [N,128]
  float*    Bb    = A  + (size_t)NN * KF;         // lin buffer    [N,128]
  float*    Cc    = Bb + (size_t)NN * KF;         // gat/x buffer  [N,128]
  float*    Dd    = Cc + (size_t)NN * KF;         // gat/x buffer  [N,128]
  float*    asrc  = Dd + (size_t)NN * KF;         // [N,4]
  float*    adst  = asrc + (size_t)NN * HH;       // [N,4]
  unsigned* menc  = (unsigned*)(adst + (size_t)NN * HH);  // [N,4]
  float*    denom = (float*)menc + (size_t)NN * HH;       // [N,4]
  float*    pbuf  = denom + (size_t)NN * HH;      // [Etot,4]

  const dim3 blk(256);
  auto cdiv = [](long long a, long long b) { return (int)((a + b - 1) / b); };
  const int gN  = cdiv(NN, 16 * RT);               // 782 blocks of 64 rows
  const int gNH = cdiv((long long)NN * HH, 256);
  const int gNF = cdiv((long long)NN * KF, 256);
  const int gEt = cdiv(ETOT, 256);
  const unsigned NEG_INF_ENC = 0x007FFFFFu;        // fenc(-inf)

  // ==================== layer 1 (128 -> 4x32, concat) ====================
  gemm_wmma_k<<<gN, blk, 0, stream>>>(x, W1, nullptr, A, KF);
  gemm_wmma_k<<<gN, blk, 0, stream>>>(x, l1W, l1b, Bb, KF);
  alpha_k<<<gNH, blk, 0, stream>>>(A, a1s, a1d, asrc, adst, 32, KF);
  fill_u32_k<<<gNH, blk, 0, stream>>>(menc, NEG_INF_ENC, NN * HH);
  fill_u32_k<<<gNH, blk, 0, stream>>>((unsigned*)denom, 0u, NN * HH);
  fill_u32_k<<<gNF, blk, 0, stream>>>((unsigned*)Cc, 0u, NN * KF);
  edge_max_k<<<gEt, blk, 0, stream>>>(ei, asrc, adst, menc, pbuf);
  edge_expsum_k<<<gEt, blk, 0, stream>>>(ei, menc, denom, pbuf);
  edge_norm_k<<<gEt, blk, 0, stream>>>(ei, denom, pbuf);
  edge_agg_k<<<ETOT, 128, 0, stream>>>(ei, pbuf, A, Cc, KF, 32);
  finalize12_k<<<gNF, blk, 0, stream>>>(Cc, b1, Bb, Cc);   // x2 in Cc

  // ==================== layer 2 (128 -> 4x32, concat) ====================
  gemm_wmma_k<<<gN, blk, 0, stream>>>(Cc, W2, nullptr, A, KF);
  gemm_wmma_k<<<gN, blk, 0, stream>>>(Cc, l2W, l2b, Bb, KF);
  alpha_k<<<gNH, blk, 0, stream>>>(A, a2s, a2d, asrc, adst, 32, KF);
  fill_u32_k<<<gNH, blk, 0, stream>>>(menc, NEG_INF_ENC, NN * HH);
  fill_u32_k<<<gNH, blk, 0, stream>>>((unsigned*)denom, 0u, NN * HH);
  fill_u32_k<<<gNF, blk, 0, stream>>>((unsigned*)Dd, 0u, NN * KF);
  edge_max_k<<<gEt, blk, 0, stream>>>(ei, asrc, adst, menc, pbuf);
  edge_expsum_k<<<gEt, blk, 0, stream>>>(ei, menc, denom, pbuf);
  edge_norm_k<<<gEt, blk, 0, stream>>>(ei, denom, pbuf);
  edge_agg_k<<<ETOT, 128, 0, stream>>>(ei, pbuf, A, Dd, KF, 32);
  finalize12_k<<<gNF, blk, 0, stream>>>(Dd, b2, Bb, Dd);   // x3 in Dd

  // ============== layer 3 (128 -> 4x10, mean over heads) =================
  gemm_wmma_k<<<gN, blk, 0, stream>>>(Dd, W3, nullptr, A, 40);   // h3  [N,40]
  gemm_wmma_k<<<gN, blk, 0, stream>>>(Dd, l3W, l3b, Bb, 10);     // lin3[N,10]
  alpha_k<<<gNH, blk, 0, stream>>>(A, a3s, a3d, asrc, adst, 10, 40);
  fill_u32_k<<<gNH, blk, 0, stream>>>(menc, NEG_INF_ENC, NN * HH);
  fill_u32_k<<<gNH, blk, 0, stream>>>((unsigned*)denom, 0u, NN * HH);
  fill_u32_k<<<cdiv((long long)NN * 40, 256), blk, 0, stream>>>((unsigned*)Cc, 0u, NN * 40);
  edge_max_k<<<gEt, blk, 0, stream>>>(ei, asrc, adst, menc, pbuf);
  edge_expsum_k<<<gEt, blk, 0, stream>>>(ei, menc, denom, pbuf);
  edge_norm_k<<<gEt, blk, 0, stream>>>(ei, denom, pbuf);
  edge_agg_k<<<ETOT, 64, 0, stream>>>(ei, pbuf, A, Cc, 40, 10);
  finalize3_k<<<cdiv((long long)NN * 10, 256), blk, 0, stream>>>(Cc, b3, Bb, out);
}